// HybnetLoss_plus_22651657519184
// MI455X (gfx1250) — compile-verified
//
#include <hip/hip_runtime.h>

// ---------------------------------------------------------------------------
// CDNA5 (gfx1250) implementation of HybnetLoss.
// Both 1024x1024 eigh calls are replaced by rank-128 linear algebra ->
// everything is f32 WMMA GEMMs plus small one-workgroup Cholesky / Jacobi
// kernels. Working set ~12 MB -> fully L2-resident (192 MB L2).
// ---------------------------------------------------------------------------

typedef float v2f __attribute__((ext_vector_type(2)));
typedef float v8f __attribute__((ext_vector_type(8)));

#define MF   128        // M_FILTERS
#define NC   1024       // N_CHANNELS
#define BATCH_N (4096 * 1024)
#define PAR_N   (128 * 17)
#define DN   (MF * NC)  // 131072
#define IMAX 50
#define SUBSPACE_STEPS 2
#define JACOBI_SWEEPS  6

// ---------------------------------------------------------------------------
// Generic f32 WMMA GEMM:  C[M,N] = alpha * A(M,K) * B(K,N)
// A element (i,k) at A[i*ars + k*acs]; B element (k,j) at B[k*brs + j*bcs].
// Each wave computes a 16x64 strip: one A fragment feeds 4 WMMA accumulators
// (4x arithmetic intensity vs one tile/wave). K-step 4 via
// V_WMMA_F32_16X16X4_F32. ISA 16x4 f32 A layout: lanes 0-15 -> K=0,1 ;
// lanes 16-31 -> K=2,3. C/D layout: VGPR r: lanes0-15 = (M=r, N=lane);
// lanes16-31 = (M=r+8, N=lane-16). Requires M%16==0, N%64==0, K%4==0.
// ---------------------------------------------------------------------------
__global__ __launch_bounds__(128) void wmma_gemm_f32(
    const float* __restrict__ A, int ars, int acs,
    const float* __restrict__ B, int brs, int bcs,
    float* __restrict__ C, int ldc,
    int M, int N, int K, float alpha)
{
    int wave = blockIdx.x * (blockDim.x >> 5) + (threadIdx.x >> 5);
    int lane = threadIdx.x & 31;
    int stripsN = N >> 6;
    int total   = (M >> 4) * stripsN;
    if (wave >= total) return;                 // whole-wave uniform: EXEC stays all-1s
    int tm = (wave / stripsN) << 4;
    int tn = (wave % stripsN) << 6;
    int l = lane & 15, h = lane >> 4;

    const float* Arow = A + (size_t)(tm + l) * ars;
    const float* B0   = B + (size_t)(tn + l) * bcs;
    const float* B1   = B0 + (size_t)16 * bcs;
    const float* B2   = B0 + (size_t)32 * bcs;
    const float* B3   = B0 + (size_t)48 * bcs;

    v8f acc0 = {}, acc1 = {}, acc2 = {}, acc3 = {};
    for (int k0 = 0; k0 < K; k0 += 4) {
        int ka = k0 + 2 * h;
        size_t o0 = (size_t)ka * acs, o1 = (size_t)(ka + 1) * acs;
        size_t r0 = (size_t)ka * brs, r1 = (size_t)(ka + 1) * brs;
        if (k0 + 8 < K)   // wave-uniform: prefetch next A K-block into L0
            __builtin_prefetch(Arow + (size_t)(ka + 8) * acs, 0, 1);
        v2f a, b0, b1, b2, b3;
        a.x  = Arow[o0]; a.y  = Arow[o1];
        b0.x = B0[r0];   b0.y = B0[r1];
        b1.x = B1[r0];   b1.y = B1[r1];
        b2.x = B2[r0];   b2.y = B2[r1];
        b3.x = B3[r0];   b3.y = B3[r1];
        // 8 args: (neg_a, A, neg_b, B, c_mod, C, reuse_a, reuse_b)
        acc0 = __builtin_amdgcn_wmma_f32_16x16x4_f32(false, a, false, b0, (short)0, acc0, false, false);
        acc1 = __builtin_amdgcn_wmma_f32_16x16x4_f32(false, a, false, b1, (short)0, acc1, false, false);
        acc2 = __builtin_amdgcn_wmma_f32_16x16x4_f32(false, a, false, b2, (short)0, acc2, false, false);
        acc3 = __builtin_amdgcn_wmma_f32_16x16x4_f32(false, a, false, b3, (short)0, acc3, false, false);
    }
    float* Crow = C + (size_t)(tm + 8 * h) * ldc + tn + l;
    #pragma unroll
    for (int r = 0; r < 8; ++r) {
        Crow[(size_t)r * ldc +  0] = alpha * acc0[r];
        Crow[(size_t)r * ldc + 16] = alpha * acc1[r];
        Crow[(size_t)r * ldc + 32] = alpha * acc2[r];
        Crow[(size_t)r * ldc + 48] = alpha * acc3[r];
    }
}

// ---------------------------------------------------------------------------
// Streaming sum((a-b)^2) -> per-block partials
// ---------------------------------------------------------------------------
__global__ __launch_bounds__(256) void sq_diff_partial(
    const float* __restrict__ a, const float* __restrict__ b,
    float* __restrict__ part, int n)
{
    __shared__ float red[256];
    float s = 0.f;
    for (int i = blockIdx.x * 256 + threadIdx.x; i < n; i += gridDim.x * 256) {
        float d = a[i] - b[i];
        s = fmaf(d, d, s);
    }
    red[threadIdx.x] = s; __syncthreads();
    for (int st = 128; st > 0; st >>= 1) {
        if ((int)threadIdx.x < st) red[threadIdx.x] += red[threadIdx.x + st];
        __syncthreads();
    }
    if (threadIdx.x == 0) part[blockIdx.x] = red[0];
}

// original_loss = mean(sq) + beta * mean(relu(range residual))
__global__ __launch_bounds__(256) void finalize_base(
    const float* __restrict__ part, int nparts,
    const float* __restrict__ params,
    const float* __restrict__ tminp, const float* __restrict__ tmaxp,
    const float* __restrict__ betap, float* __restrict__ scal)
{
    __shared__ float r1[256], r2[256];
    float s = 0.f, r = 0.f;
    for (int i = threadIdx.x; i < nparts; i += 256) s += part[i];
    float tmin = *tminp, tmax = *tmaxp;
    const float d = 0.01f;
    for (int i = threadIdx.x; i < PAR_N; i += 256) {
        float p = params[i];
        float res = fmaxf((p - tmin - d) / -d, (p - tmax + d) / d);
        r += fmaxf(res, 0.f);
    }
    r1[threadIdx.x] = s; r2[threadIdx.x] = r; __syncthreads();
    for (int st = 128; st > 0; st >>= 1) {
        if ((int)threadIdx.x < st) {
            r1[threadIdx.x] += r1[threadIdx.x + st];
            r2[threadIdx.x] += r2[threadIdx.x + st];
        }
        __syncthreads();
    }
    if (threadIdx.x == 0)
        scal[0] = r1[0] / (float)BATCH_N + (*betap) * r2[0] / (float)PAR_N;
}

__global__ void init_iter(unsigned* rthbits) { *rthbits = 0x7F7FFFFFu; }

// Ri[j] = sum_i |G[j,i]| - |G[j,j]|  (G symmetric -> row sum == column sum,
// row read is coalesced);  Rth = min_j Ri[j] via monotone uint atomicMin.
__global__ __launch_bounds__(256) void ri_kernel(
    const float* __restrict__ G, float* __restrict__ Ri, unsigned* __restrict__ rthbits)
{
    __shared__ float red[256];
    int j = blockIdx.x;
    float s = 0.f;
    for (int i = threadIdx.x; i < NC; i += 256) s += fabsf(G[(size_t)j * NC + i]);
    red[threadIdx.x] = s; __syncthreads();
    for (int st = 128; st > 0; st >>= 1) {
        if ((int)threadIdx.x < st) red[threadIdx.x] += red[threadIdx.x + st];
        __syncthreads();
    }
    if (threadIdx.x == 0) {
        float r = red[0] - fabsf(G[(size_t)j * NC + j]);
        Ri[j] = r;
        atomicMin(rthbits, __float_as_uint(r));   // r >= 0 -> bit order monotone
    }
}

// Gn = where(Ri[i]>Rth, delta*sign(G), G) with diag kept; G2 = (Gn + Gn^T)/2
__global__ __launch_bounds__(256) void gn_sym_kernel(
    const float* __restrict__ G, const float* __restrict__ Ri,
    const unsigned* __restrict__ rthbits, float* __restrict__ G2)
{
    size_t idx = (size_t)blockIdx.x * 256 + threadIdx.x;   // 1M entries
    int i = (int)(idx >> 10), j = (int)(idx & 1023);
    float rth = __uint_as_float(*rthbits);
    float delta = rth / (float)(NC - 1);
    float gij = G[idx];
    float gji = G[(size_t)j * NC + i];
    float a = (i != j && Ri[i] > rth) ? delta * (float)((gij > 0.f) - (gij < 0.f)) : gij;
    float b = (i != j && Ri[j] > rth) ? delta * (float)((gji > 0.f) - (gji < 0.f)) : gji;
    G2[idx] = 0.5f * (a + b);
}

// ---------------------------------------------------------------------------
// One-workgroup SPD Cholesky A = L L^T, then Minv = L^{-1} (lower tri).
// Dynamic LDS: 2 * 128*128 f32 = 128 KB (WGP supports 320 KB).
// ---------------------------------------------------------------------------
__global__ __launch_bounds__(128) void chol_inv_kernel(
    const float* __restrict__ Ain, float* __restrict__ Minv)
{
    extern __shared__ float lds[];
    float* a = lds;              // L (in place)
    float* m = lds + MF * MF;    // L^{-1}
    int tid = threadIdx.x;
    for (int i = tid; i < MF * MF; i += 128) a[i] = Ain[i];
    __syncthreads();
    for (int k = 0; k < MF; ++k) {
        if (tid == 0) a[k * MF + k] = sqrtf(fmaxf(a[k * MF + k], 1e-30f));
        __syncthreads();
        if (tid > k) a[tid * MF + k] /= a[k * MF + k];
        __syncthreads();
        int w = MF - 1 - k;
        for (int idx = tid; idx < w * w; idx += 128) {
            int j = k + 1 + idx / w;
            int i = k + 1 + idx % w;
            if (i >= j) a[i * MF + j] -= a[i * MF + k] * a[j * MF + k];
        }
        __syncthreads();
    }
    // column-parallel forward substitution: thread j builds column j of L^{-1}
    if (tid < MF) {
        int j = tid;
        for (int i = 0; i < j; ++i) m[i * MF + j] = 0.f;
        m[j * MF + j] = 1.f / a[j * MF + j];
        for (int i = j + 1; i < MF; ++i) {
            float s = 0.f;
            for (int k = j; k < i; ++k) s += a[i * MF + k] * m[k * MF + j];
            m[i * MF + j] = -s / a[i * MF + i];
        }
    }
    __syncthreads();
    for (int i = tid; i < MF * MF; i += 128) Minv[i] = m[i];
}

// ---------------------------------------------------------------------------
// Parallel cyclic (tournament) Jacobi on 128x128 symmetric H.
// Outputs eigenvectors W (columns), eigenvalues w, descending order ord.
// Dynamic LDS: H + W = 128 KB.
// ---------------------------------------------------------------------------
__global__ __launch_bounds__(256) void jacobi_kernel(
    const float* __restrict__ Hin, float* __restrict__ Wout,
    float* __restrict__ wvals, int* __restrict__ ord, int sweeps)
{
    extern __shared__ float lds[];
    float* H = lds;
    float* W = lds + MF * MF;
    __shared__ int   perm[MF];
    __shared__ float cs[64], sn[64];
    int tid = threadIdx.x;
    for (int i = tid; i < MF * MF; i += 256) {
        H[i] = Hin[i];
        W[i] = ((i >> 7) == (i & 127)) ? 1.f : 0.f;
    }
    if (tid < MF) perm[tid] = tid;
    __syncthreads();

    for (int sw = 0; sw < sweeps; ++sw) {
        for (int round = 0; round < MF - 1; ++round) {
            if (tid < 64) {
                int p = perm[tid], q = perm[MF - 1 - tid];
                if (p > q) { int t = p; p = q; q = t; }
                float app = H[p * MF + p], aqq = H[q * MF + q], apq = H[p * MF + q];
                float c = 1.f, s = 0.f;
                if (fabsf(apq) > 1e-12f) {
                    float tau = (aqq - app) / (2.f * apq);
                    float t = (tau >= 0.f) ? 1.f / (tau + sqrtf(1.f + tau * tau))
                                           : 1.f / (tau - sqrtf(1.f + tau * tau));
                    c = rsqrtf(1.f + t * t); s = t * c;
                }
                cs[tid] = c; sn[tid] = s;
            }
            __syncthreads();
            // column rotations (disjoint column pairs) on H and W
            for (int idx = tid; idx < 64 * MF; idx += 256) {
                int pr = idx >> 7, k = idx & 127;
                int p = perm[pr], q = perm[MF - 1 - pr];
                if (p > q) { int t = p; p = q; q = t; }
                float c = cs[pr], s = sn[pr];
                float hp = H[k * MF + p], hq = H[k * MF + q];
                H[k * MF + p] = c * hp - s * hq;
                H[k * MF + q] = s * hp + c * hq;
                float wp = W[k * MF + p], wq = W[k * MF + q];
                W[k * MF + p] = c * wp - s * wq;
                W[k * MF + q] = s * wp + c * wq;
            }
            __syncthreads();
            // row rotations on H
            for (int idx = tid; idx < 64 * MF; idx += 256) {
                int pr = idx >> 7, k = idx & 127;
                int p = perm[pr], q = perm[MF - 1 - pr];
                if (p > q) { int t = p; p = q; q = t; }
                float c = cs[pr], s = sn[pr];
                float hp = H[p * MF + k], hq = H[q * MF + k];
                H[p * MF + k] = c * hp - s * hq;
                H[q * MF + k] = s * hp + c * hq;
            }
            __syncthreads();
            if (tid == 0) {   // rotate tournament schedule (fix player 0)
                int last = perm[MF - 1];
                for (int i = MF - 1; i > 1; --i) perm[i] = perm[i - 1];
                perm[1] = last;
            }
            __syncthreads();
        }
    }
    if (tid < MF) wvals[tid] = H[tid * MF + tid];
    for (int i = tid; i < MF * MF; i += 256) Wout[i] = W[i];
    __syncthreads();
    if (tid == 0) {   // descending order (reads diag from LDS)
        unsigned taken[4] = {0, 0, 0, 0};
        for (int m = 0; m < MF; ++m) {
            int best = 0; float bv = -3.4e38f;
            for (int j = 0; j < MF; ++j) {
                if (taken[j >> 5] & (1u << (j & 31))) continue;
                float v = H[j * MF + j];
                if (v > bv) { bv = v; best = j; }
            }
            taken[best >> 5] |= 1u << (best & 31);
            ord[m] = best;
        }
    }
}

// Dnew[m, i] = sqrt(max(w[ord[m]],0)) * U[i, ord[m]]
__global__ __launch_bounds__(256) void scale_store_D(
    const float* __restrict__ U, const float* __restrict__ w,
    const int* __restrict__ ord, float* __restrict__ Dn)
{
    int idx = blockIdx.x * 256 + threadIdx.x;   // DN entries
    int m = idx >> 10, i = idx & 1023;
    int o = ord[m];
    Dn[idx] = sqrtf(fmaxf(w[o], 0.f)) * U[(size_t)i * MF + o];
}

__global__ __launch_bounds__(256) void final_combine(
    const float* __restrict__ part, int nparts,
    const float* __restrict__ scal, float* __restrict__ out)
{
    __shared__ float red[256];
    float s = 0.f;
    for (int i = threadIdx.x; i < nparts; i += 256) s += part[i];
    red[threadIdx.x] = s; __syncthreads();
    for (int st = 128; st > 0; st >>= 1) {
        if ((int)threadIdx.x < st) red[threadIdx.x] += red[threadIdx.x + st];
        __syncthreads();
    }
    if (threadIdx.x == 0) out[0] = scal[0] + red[0] / (float)DN;
}

// ---------------------------------------------------------------------------
static inline void gemm(hipStream_t st,
                        const float* A, int ars, int acs,
                        const float* B, int brs, int bcs,
                        float* C, int ldc, int M, int N, int K, float alpha)
{
    int waves = (M / 16) * (N / 64);           // one 16x64 strip per wave
    int blocks = (waves + 3) / 4;
    wmma_gemm_f32<<<blocks, 128, 0, st>>>(A, ars, acs, B, brs, bcs, C, ldc, M, N, K, alpha);
}

extern "C" void kernel_launch(void* const* d_in, const int* in_sizes, int n_in,
                              void* d_out, int out_size, void* d_ws, size_t ws_size,
                              hipStream_t stream)
{
    const float* t1     = (const float*)d_in[0];
    const float* t2     = (const float*)d_in[1];
    const float* params = (const float*)d_in[2];
    const float* tmin   = (const float*)d_in[3];
    const float* tmax   = (const float*)d_in[4];
    const float* beta   = (const float*)d_in[5];
    const float* resp   = (const float*)d_in[6];
    float* out = (float*)d_out;

    // workspace arena (~12.3 MB of f32)
    float* ws   = (float*)d_ws;
    float* SCAL = ws;                 // [0]=base loss, [1]=Rth bits
    float* PART = ws + 16;            // up to 1024 partials
    float* D0   = PART + 1024;
    float* D1   = D0 + DN;
    float* G    = D1 + DN;
    float* G2   = G + (size_t)NC * NC;
    float* T    = G2 + (size_t)NC * NC;
    float* SA   = T + DN;             // 128x128 scratch (Gram / A)
    float* SM   = SA + MF * MF;       // L^{-1}
    float* SW   = SM + MF * MF;       // Jacobi eigenvectors
    float* SH   = SW + MF * MF;       // Ainv / Ritz matrix
    float* EVW  = SH + MF * MF;       // 128 eigenvalues
    int*   ORD  = (int*)(EVW + MF);   // 128 order indices
    float* RI   = (float*)(ORD + MF); // 1024 row sums
    float* X    = RI + NC;            // 1024x128 subspace
    float* Y    = X + DN;
    float* Z    = Y + DN;
    float* U    = Z + DN;
    (void)in_sizes; (void)n_in; (void)out_size; (void)ws_size;

    const size_t shsmall = (size_t)2 * MF * MF * sizeof(float);   // 128 KB LDS

    // base losses + D init
    hipMemcpyAsync(D0, resp, DN * sizeof(float), hipMemcpyDeviceToDevice, stream);
    sq_diff_partial<<<1024, 256, 0, stream>>>(t1, t2, PART, BATCH_N);
    finalize_base<<<1, 256, 0, stream>>>(PART, 1024, params, tmin, tmax, beta, SCAL);

    float* Dc = D0; float* Dn = D1;
    for (int it = 0; it < IMAX; ++it) {
        // --- first eigh == rank-128 projector: G = ratio * D^T (D D^T)^-1 D
        gemm(stream, Dc, NC, 1, Dc, 1, NC, SA, MF, MF, MF, NC, 1.0f);      // A = D D^T
        chol_inv_kernel<<<1, 128, shsmall, stream>>>(SA, SM);              // SM = L^-1
        gemm(stream, SM, 1, MF, SM, MF, 1, SH, MF, MF, MF, MF, 1.0f);      // Ainv = SM^T SM
        gemm(stream, SH, MF, 1, Dc, NC, 1, T, NC, MF, NC, MF, 1.0f);       // T = Ainv D
        gemm(stream, Dc, 1, NC, T, NC, 1, G, NC, NC, NC, MF, 8.0f);        // G = ratio D^T T

        // --- coherence clipping + symmetrization
        init_iter<<<1, 1, 0, stream>>>((unsigned*)(SCAL + 1));
        ri_kernel<<<NC, 256, 0, stream>>>(G, RI, (unsigned*)(SCAL + 1));
        gn_sym_kernel<<<(NC * NC) / 256, 256, 0, stream>>>(G, RI, (unsigned*)(SCAL + 1), G2);

        // --- second eigh: warm-started subspace iteration + Rayleigh-Ritz
        for (int s = 0; s < SUBSPACE_STEPS; ++s) {
            if (s == 0)
                gemm(stream, G2, NC, 1, Dc, 1, NC, Y, MF, NC, MF, NC, 1.0f); // Y = G2 D^T
            else
                gemm(stream, G2, NC, 1, X, MF, 1, Y, MF, NC, MF, NC, 1.0f);  // Y = G2 X
            gemm(stream, Y, 1, MF, Y, MF, 1, SA, MF, MF, MF, NC, 1.0f);      // B = Y^T Y
            chol_inv_kernel<<<1, 128, shsmall, stream>>>(SA, SM);            // SM = L^-1
            gemm(stream, Y, MF, 1, SM, 1, MF, X, MF, NC, MF, MF, 1.0f);      // X = Y L^-T
        }
        gemm(stream, G2, NC, 1, X, MF, 1, Z, MF, NC, MF, NC, 1.0f);          // Z = G2 X
        gemm(stream, X, 1, MF, Z, MF, 1, SH, MF, MF, MF, NC, 1.0f);          // H = X^T Z
        jacobi_kernel<<<1, 256, shsmall, stream>>>(SH, SW, EVW, ORD, JACOBI_SWEEPS);
        gemm(stream, X, MF, 1, SW, MF, 1, U, MF, NC, MF, MF, 1.0f);          // U = X W
        scale_store_D<<<DN / 256, 256, 0, stream>>>(U, EVW, ORD, Dn);

        float* tmp = Dc; Dc = Dn; Dn = tmp;
    }

    // rloss = mean((D - responses)^2); out = original_loss + rloss
    sq_diff_partial<<<64, 256, 0, stream>>>(Dc, resp, PART, DN);
    final_combine<<<1, 256, 0, stream>>>(PART, 64, SCAL, out);
}